// ToRGBLayer_30511447671560
// MI455X (gfx1250) — compile-verified
//
#include <hip/hip_runtime.h>

typedef __attribute__((ext_vector_type(2))) float v2f;
typedef __attribute__((ext_vector_type(8))) float v8f;

#define B_SZ    4
#define C_IN    512
#define C_OUT   96
#define W_DIM   512
#define SPATIAL 32768   // 32^3
#define N_TILE  16
#define M_TILES 6       // 96 / 16
#define KP      64      // K panel staged in LDS
#define PADM    112     // padded M stride in float2 units: 112*2 % 64 == 32 -> lane halves hit disjoint bank sets

// ---------------------------------------------------------------------------
// Kernel 1: styles[b][c] = ((w[b,:] . affine_w[c,:]) / sqrt(512) + bias[c]) / sqrt(512)
// (second factor is ToRGB weight_gain = k^3 / sqrt(C_in) with k = 1)
// ---------------------------------------------------------------------------
__global__ void styles_kernel(const float* __restrict__ w,
                              const float* __restrict__ affine_w,
                              const float* __restrict__ affine_b,
                              float* __restrict__ styles) {
    int tid = blockIdx.x * blockDim.x + threadIdx.x;
    if (tid >= B_SZ * C_IN) return;
    int b = tid / C_IN;
    int c = tid % C_IN;
    const float* wr = w + b * W_DIM;
    const float* ar = affine_w + c * W_DIM;
    float acc = 0.f;
    for (int j = 0; j < W_DIM; ++j) acc = fmaf(wr[j], ar[j], acc);
    const float inv_sqrt512 = 0.04419417382415922f;  // 1/sqrt(512)
    float s = acc * inv_sqrt512 + affine_b[c];
    styles[tid] = s * inv_sqrt512;
}

// ---------------------------------------------------------------------------
// Kernel 2: fold styles into the conv weight, stored K-major (transposed):
//   wmodT[b][k][m] = weight[m][k] * styles[b][k]     (786 KB total)
// ---------------------------------------------------------------------------
__global__ void wmod_kernel(const float* __restrict__ weight,
                            const float* __restrict__ styles,
                            float* __restrict__ wmodT) {
    int tid = blockIdx.x * blockDim.x + threadIdx.x;
    if (tid >= B_SZ * C_IN * C_OUT) return;
    int m = tid % C_OUT;
    int k = (tid / C_OUT) % C_IN;
    int b = tid / (C_OUT * C_IN);
    wmodT[tid] = weight[m * C_IN + k] * styles[b * C_IN + k];
}

// ---------------------------------------------------------------------------
// Kernel 3: per batch, out[96 x 32768] = WmodT^T [96x512] @ x [512x32768] + bias
// using V_WMMA_F32_16X16X4_F32. One wave = one 16-wide N strip x all 6 M tiles.
// 8 waves/block share the LDS-staged WmodT K-panel (same batch).
// LDS holds the panel as float2{row k, row k+1} per m, so each A fragment is
// one aligned ds_load_b64 straight into an even VGPR pair (no repack movs).
// ---------------------------------------------------------------------------
__global__ __launch_bounds__(256) void torgb_gemm_kernel(
    const float* __restrict__ x,      // [B][C_IN][SPATIAL]
    const float* __restrict__ wmodT,  // [B][C_IN][C_OUT]  (K-major)
    const float* __restrict__ bias,   // [C_OUT]
    float* __restrict__ out)          // [B][C_OUT][SPATIAL]
{
    __shared__ v2f ldsA[(KP / 2) * PADM];   // 28 KB

    const int b    = blockIdx.y;
    const int wave = threadIdx.x >> 5;
    const int lane = threadIdx.x & 31;
    const int half = lane >> 4;       // 0: lanes 0-15, 1: lanes 16-31
    const int lm   = lane & 15;
    const int nbase = (blockIdx.x * 8 + wave) * N_TILE;

    const float* xb = x     + (size_t)b * C_IN * SPATIAL;
    const float* wb = wmodT + (size_t)b * C_IN * C_OUT;

    v8f acc[M_TILES];
    v8f zero = {};
    #pragma unroll
    for (int t = 0; t < M_TILES; ++t) acc[t] = zero;

    for (int kp = 0; kp < C_IN; kp += KP) {
        __syncthreads();
        // Cooperative, coalesced global read of A panel [KP][C_OUT];
        // scatter into LDS with the (k, k+1) pair interleaved per m.
        for (int i = threadIdx.x; i < KP * C_OUT; i += 256) {
            int kk = i / C_OUT;
            int mm = i % C_OUT;
            ((float*)ldsA)[(((kk >> 1) * PADM) + mm) * 2 + (kk & 1)] =
                wb[(size_t)(kp + kk) * C_OUT + mm];
        }
        __syncthreads();

        for (int k = 0; k < KP; k += 4) {
            const int krow = k + 2 * half;   // this lane-half's K rows (krow, krow+1)

            // B fragment: lane holds x[b][krow+v][nbase+lm]  (v = vgpr index)
            v2f bfrag;
            const float* xp = xb + (size_t)(kp + krow) * SPATIAL + nbase + lm;
            bfrag.x = xp[0];
            bfrag.y = xp[SPATIAL];

            // A fragments: one b64 load each, kpair = krow/2
            const v2f* ap = ldsA + ((k >> 1) + half) * PADM + lm;
            #pragma unroll
            for (int t = 0; t < M_TILES; ++t) {
                v2f afrag = ap[t * 16];
                acc[t] = __builtin_amdgcn_wmma_f32_16x16x4_f32(
                    /*neg_a=*/false, afrag, /*neg_b=*/false, bfrag,
                    /*c_mod=*/(short)0, acc[t],
                    /*reuse_a=*/false, /*reuse_b=*/false);
            }
        }
    }

    // C/D layout: VGPR r, lane-half h -> row m = t*16 + r + 8*h, col n = nbase + lm
    #pragma unroll
    for (int t = 0; t < M_TILES; ++t) {
        #pragma unroll
        for (int r = 0; r < 8; ++r) {
            int m = t * 16 + r + 8 * half;
            out[((size_t)b * C_OUT + m) * SPATIAL + nbase + lm] = acc[t][r] + bias[m];
        }
    }
}

// ---------------------------------------------------------------------------
extern "C" void kernel_launch(void* const* d_in, const int* in_sizes, int n_in,
                              void* d_out, int out_size, void* d_ws, size_t ws_size,
                              hipStream_t stream) {
    const float* x    = (const float*)d_in[0];  // [4,512,32,32,32]
    const float* w    = (const float*)d_in[1];  // [4,512]
    const float* aw   = (const float*)d_in[2];  // [512,512]
    const float* ab   = (const float*)d_in[3];  // [512]
    const float* wt   = (const float*)d_in[4];  // [96,512,1,1,1]
    const float* bias = (const float*)d_in[5];  // [96]
    float* out = (float*)d_out;

    float* styles = (float*)d_ws;                 // 4*512 floats
    float* wmodT  = styles + B_SZ * C_IN;         // 4*512*96 floats

    styles_kernel<<<(B_SZ * C_IN + 255) / 256, 256, 0, stream>>>(w, aw, ab, styles);
    wmod_kernel<<<(B_SZ * C_IN * C_OUT + 255) / 256, 256, 0, stream>>>(wt, styles, wmodT);

    dim3 grid(SPATIAL / N_TILE / 8, B_SZ);        // (256, 4) blocks of 256 threads
    torgb_gemm_kernel<<<grid, 256, 0, stream>>>(x, wmodT, bias, out);
}